// CaptionHead_58832462021206
// MI455X (gfx1250) — compile-verified
//
#include <hip/hip_runtime.h>
#include <math.h>

// ---------------------------------------------------------------------------
// CaptionHead pipeline for MI455X (gfx1250, wave32)
//   sums[s]  = sum_t adapter[v2p[pidx[t]]]   (seg_ids[t]==s)
//   pooledN  = l2norm(sums / max(count,1));  ceN = l2norm(caption_embed)
//   out      = (pooledN @ ceN^T) * exp(logit_scale)      [S,S] f32
// ---------------------------------------------------------------------------

#define D_DIM 256

typedef __attribute__((ext_vector_type(2))) float v2f;
typedef __attribute__((ext_vector_type(8))) float v8f;

// ---- kernel 0: zero the accumulators in workspace -------------------------
__global__ void ch_zero(float* __restrict__ p, int n) {
    int i = blockIdx.x * blockDim.x + threadIdx.x;
    int stride = gridDim.x * blockDim.x;
    for (; i < n; i += stride) p[i] = 0.0f;
}

// ---- kernel 1: composed gather + segment accumulation ---------------------
// One wave (32 lanes) per row t: lane L owns 8 consecutive floats of the
// 256-wide feature row (two b128 loads), then 8 f32 L2 atomics.
__global__ void ch_seg_accum(const float* __restrict__ adapter,
                             const int*   __restrict__ v2p,
                             const int*   __restrict__ pidx,
                             const int*   __restrict__ segs,
                             float*       __restrict__ sums,
                             float*       __restrict__ counts,
                             int T) {
    const int lane = threadIdx.x & 31;
    const int wave = threadIdx.x >> 5;
    const int t = blockIdx.x * (blockDim.x >> 5) + wave;
    if (t >= T) return;

    const int s = segs[t];
    const int v = v2p[pidx[t]];

    const float4* src = (const float4*)(adapter + (size_t)v * D_DIM) + lane * 2;
    float4 r0 = src[0];
    float4 r1 = src[1];

    float* dst = sums + (size_t)s * D_DIM + lane * 8;
    unsafeAtomicAdd(dst + 0, r0.x);
    unsafeAtomicAdd(dst + 1, r0.y);
    unsafeAtomicAdd(dst + 2, r0.z);
    unsafeAtomicAdd(dst + 3, r0.w);
    unsafeAtomicAdd(dst + 4, r1.x);
    unsafeAtomicAdd(dst + 5, r1.y);
    unsafeAtomicAdd(dst + 6, r1.z);
    unsafeAtomicAdd(dst + 7, r1.w);
    if (lane == 0) unsafeAtomicAdd(&counts[s], 1.0f);
}

// ---- kernel 2: mean + L2-normalize (pooled rows and caption rows) ---------
// row in [0,S)   -> pooledN[row] = l2norm(sums[row] / max(counts,1))
// row in [S,2S)  -> ceN[row-S]   = l2norm(caption_embed[row-S])
__global__ void ch_normalize(const float* __restrict__ sums,
                             const float* __restrict__ counts,
                             const float* __restrict__ cap,
                             float*       __restrict__ pooledN,
                             float*       __restrict__ ceN,
                             int S) {
    const int lane = threadIdx.x & 31;
    const int wave = threadIdx.x >> 5;
    const int row = blockIdx.x * (blockDim.x >> 5) + wave;
    if (row >= 2 * S) return;

    const bool isPooled = row < S;
    const int  s = isPooled ? row : row - S;
    const float* src = (isPooled ? sums : cap) + (size_t)s * D_DIM;

    float inv_c = 1.0f;
    if (isPooled) inv_c = 1.0f / fmaxf(counts[s], 1.0f);

    float4 a = ((const float4*)src)[lane * 2 + 0];
    float4 b = ((const float4*)src)[lane * 2 + 1];
    a.x *= inv_c; a.y *= inv_c; a.z *= inv_c; a.w *= inv_c;
    b.x *= inv_c; b.y *= inv_c; b.z *= inv_c; b.w *= inv_c;

    float ss = a.x * a.x + a.y * a.y + a.z * a.z + a.w * a.w
             + b.x * b.x + b.y * b.y + b.z * b.z + b.w * b.w;
    // wave32 butterfly reduction
    for (int m = 16; m >= 1; m >>= 1) ss += __shfl_xor(ss, m, 32);

    const float inv = 1.0f / fmaxf(sqrtf(ss), 1e-12f);
    a.x *= inv; a.y *= inv; a.z *= inv; a.w *= inv;
    b.x *= inv; b.y *= inv; b.z *= inv; b.w *= inv;

    float4* dst = (float4*)((isPooled ? pooledN : ceN) + (size_t)s * D_DIM) + lane * 2;
    dst[0] = a;
    dst[1] = b;
}

// ---- kernel 3: logits = (pooledN @ ceN^T) * exp(logit_scale) --------------
// One wave computes one 16x16 output tile via V_WMMA_F32_16X16X4_F32.
// A fragment 16x4 (MxK): lanes 0-15 -> M=lane, K={k,k+1}; lanes 16-31 -> K={k+2,k+3}
// B fragment 4x16 (KxN): lanes 0-15 -> N=lane, K={k,k+1}; lanes 16-31 -> K={k+2,k+3}
// C VGPR r: lanes 0-15 -> M=r, N=lane; lanes 16-31 -> M=r+8, N=lane-16
__global__ void ch_wmma_gemm(const float* __restrict__ A,     // pooledN [S, 256]
                             const float* __restrict__ B,     // ceN     [S, 256]
                             const float* __restrict__ logit_scale,
                             float*       __restrict__ out,   // [S, S]
                             int S) {
    const int lane  = threadIdx.x & 31;
    const int wave  = threadIdx.x >> 5;
    const int tiles = S >> 4;                       // tiles per dim
    const int tile  = blockIdx.x * (blockDim.x >> 5) + wave;
    if (tile >= tiles * tiles) return;

    const int tm = (tile / tiles) << 4;
    const int tn = (tile % tiles) << 4;

    const int half = lane >> 4;                     // 0: K pair {0,1}; 1: {2,3}
    const int lidx = lane & 15;                     // M (for A) / N (for B)

    // both operand rows are 8-byte aligned (even float offset) -> float2 loads
    const float2* a2 = (const float2*)(A + (size_t)(tm + lidx) * D_DIM + half * 2);
    const float2* b2 = (const float2*)(B + (size_t)(tn + lidx) * D_DIM + half * 2);

    v8f c = {};
#pragma unroll 4
    for (int k = 0; k < D_DIM; k += 4) {
        float2 af = a2[k >> 1];
        float2 bf = b2[k >> 1];
        v2f a; a.x = af.x; a.y = af.y;
        v2f b; b.x = bf.x; b.y = bf.y;
        // D = A(16x4) x B(4x16) + C, full fp32
        c = __builtin_amdgcn_wmma_f32_16x16x4_f32(
                /*neg_a=*/false, a, /*neg_b=*/false, b,
                /*c_mod=*/(short)0, c, /*reuse_a=*/false, /*reuse_b=*/false);
    }

    const float scale = expf(logit_scale[0]);
    const int col   = tn + lidx;
    const int rbase = tm + half * 8;
#pragma unroll
    for (int r = 0; r < 8; ++r)
        out[(size_t)(rbase + r) * S + col] = c[r] * scale;
}

// ---------------------------------------------------------------------------
extern "C" void kernel_launch(void* const* d_in, const int* in_sizes, int n_in,
                              void* d_out, int out_size, void* d_ws, size_t ws_size,
                              hipStream_t stream) {
    const float* adapter     = (const float*)d_in[0];   // [N_VOX, 256]
    const float* caption     = (const float*)d_in[1];   // [S, 256]
    const float* logit_scale = (const float*)d_in[2];   // [1]
    const int*   v2p_map     = (const int*)  d_in[3];   // [N_PTS]
    const int*   point_idx   = (const int*)  d_in[4];   // [T]
    const int*   seg_ids     = (const int*)  d_in[5];   // [T]
    float*       out         = (float*)d_out;           // [S, S]

    const int S = in_sizes[1] / D_DIM;                  // 2048
    const int T = in_sizes[4];                          // 1e6

    // workspace layout (floats): sums[S*D] | counts[S] | pooledN[S*D] | ceN[S*D]
    float* sums    = (float*)d_ws;
    float* counts  = sums + (size_t)S * D_DIM;
    float* pooledN = counts + S;
    float* ceN     = pooledN + (size_t)S * D_DIM;

    // 0) zero accumulators
    {
        int n = S * D_DIM + S;
        int blocks = (n + 255) / 256;
        if (blocks > 4096) blocks = 4096;
        ch_zero<<<blocks, 256, 0, stream>>>(sums, n);
    }
    // 1) gather + segment accumulate (8 waves / block, 1 row / wave)
    {
        int blocks = (T + 7) / 8;
        ch_seg_accum<<<blocks, 256, 0, stream>>>(adapter, v2p_map, point_idx,
                                                 seg_ids, sums, counts, T);
    }
    // 2) normalize pooled + caption rows (2S rows, 1 row / wave)
    {
        int blocks = (2 * S + 7) / 8;
        ch_normalize<<<blocks, 256, 0, stream>>>(sums, counts, caption,
                                                 pooledN, ceN, S);
    }
    // 3) WMMA f32 GEMM with scale (one 16x16 tile / wave)
    {
        int tiles  = (S >> 4) * (S >> 4);
        int blocks = (tiles + 7) / 8;
        ch_wmma_gemm<<<blocks, 256, 0, stream>>>(pooledN, ceN, logit_scale, out, S);
    }
}